// TGGATLightweight_21638045237645
// MI455X (gfx1250) — compile-verified
//
#include <hip/hip_runtime.h>
#include <hip/hip_bf16.h>
#include <math.h>

#define N_NODES   50000
#define N_EDGES   800000
#define NDIM      128
#define HEADS     4
#define HEAD_DIM  32
#define N_GRAPHS  64

typedef __attribute__((ext_vector_type(2))) float v2f;
typedef __attribute__((ext_vector_type(8))) float v8f;
typedef __attribute__((__vector_size__(4 * sizeof(int)))) int v4i;

#define AS1 __attribute__((address_space(1)))
#define AS3 __attribute__((address_space(3)))

#if defined(__has_builtin)
#if __has_builtin(__builtin_amdgcn_global_load_async_to_lds_b128) && \
    __has_builtin(__builtin_amdgcn_s_wait_asynccnt)
#define USE_ASYNC_LDS 1
#endif
#endif
#ifndef USE_ASYNC_LDS
#define USE_ASYNC_LDS 0
#endif

// ---------------------------------------------------------------------------
// float atomic max via signed/unsigned integer atomics (IEEE-754 order trick)
// ---------------------------------------------------------------------------
__device__ __forceinline__ void atomicMaxFloat(float* addr, float val) {
    if (val >= 0.0f) {
        atomicMax((int*)addr, __float_as_int(val));
    } else {
        atomicMin((unsigned int*)addr, __float_as_uint(val));
    }
}

// ---------------------------------------------------------------------------
// WMMA fp32 GEMM:  C[M x 128] = A[M x 128] * B[128 x 128] (+ bias)
// One wave computes a 16x128 strip via 8 accumulator tiles of 16x16,
// K-loop in steps of 4 using v_wmma_f32_16x16x4_f32.
//
// LDS layout (dynamic, 135168 B per WG):
//   Bs[128 * 132]      : B stored COLUMN-major (Bs[n*132 + k]) so a B fragment
//                        (B[kb][n], B[kb+1][n]) is one aligned ds_load_b64.
//   As[8][16 * 132]    : per-wave A tile, row-major padded (As[m*132 + k]);
//                        an A fragment is one aligned ds_load_b64.
// Stride 132 (= 4 mod 64 banks) makes both fragment reads bank-conflict-free
// across the two half-wave K groups (banks 4*idx+kb vs 4*idx+kb+2).
// ---------------------------------------------------------------------------
#define BSTRIDE 132
#define ASTRIDE 132
#define GEMM_LDS_BYTES ((128 * BSTRIDE + 8 * 16 * ASTRIDE) * 4)

__global__ __launch_bounds__(256) void wmma_gemm128(
    const float* __restrict__ A, const float* __restrict__ B,
    const float* __restrict__ bias, float* __restrict__ C, int m_tiles) {
    extern __shared__ float smem[];
    float* Bs = smem;                                   // 128*132 floats
    const int wib  = threadIdx.x >> 5;                  // wave in block (0..7)
    const int lane = threadIdx.x & 31;
    const int tile = blockIdx.x * 8 + wib;              // 16-row node tile
    float* As = smem + 128 * BSTRIDE + wib * (16 * ASTRIDE);

    const int  row0   = tile * 16;
    const bool active = tile < m_tiles;                 // wave-uniform

    // ---- stage this wave's 16x128 A tile into LDS (contiguous copy) ----
    if (active) {
#if USE_ASYNC_LDS
#pragma unroll
        for (int r = 0; r < 16; ++r) {
            const float* gp = A + (size_t)(row0 + r) * NDIM + 4 * lane;
            float*       lp = As + r * ASTRIDE + 4 * lane;
            __builtin_amdgcn_global_load_async_to_lds_b128(
                (AS1 v4i*)(void*)gp, (AS3 v4i*)(void*)lp, 0, 0);
        }
#else
#pragma unroll
        for (int r = 0; r < 16; ++r) {
            const float4 v = *(const float4*)(A + (size_t)(row0 + r) * NDIM + 4 * lane);
            *(float4*)(As + r * ASTRIDE + 4 * lane) = v;
        }
#endif
    }

    // ---- cooperative stage of B into LDS, transposed to column-major ----
    for (int i = threadIdx.x; i < 128 * 128; i += 256) {
        const int k = i >> 7;        // coalesced global read along n
        const int n = i & 127;
        Bs[n * BSTRIDE + k] = B[i];
    }

#if USE_ASYNC_LDS
    __builtin_amdgcn_s_wait_asynccnt(0);
#endif
    __syncthreads();

    if (!active) return;                                // wave-uniform: EXEC full

    const int m     = lane & 15;                        // A row / B,D column
    const int khalf = lane >> 4;                        // 0: K={0,1}, 1: K={2,3}
    const float* arow = As + m * ASTRIDE;

    v8f acc[8];
#pragma unroll
    for (int t = 0; t < 8; ++t) acc[t] = {};

#pragma unroll 4
    for (int k = 0; k < 128; k += 4) {
        const int kb = k + 2 * khalf;
        const v2f a = *(const v2f*)(arow + kb);         // ds_load_b64
#pragma unroll
        for (int t = 0; t < 8; ++t) {
            const int n = t * 16 + m;
            const v2f b = *(const v2f*)(Bs + n * BSTRIDE + kb);  // ds_load_b64
            acc[t] = __builtin_amdgcn_wmma_f32_16x16x4_f32(
                false, a, false, b, (short)0, acc[t], false, false);
        }
    }

    // D layout: VGPR v, lanes 0-15 -> M=v, lanes 16-31 -> M=v+8; N = lane&15
#pragma unroll
    for (int t = 0; t < 8; ++t) {
        const int   n  = t * 16 + m;
        const float bv = bias ? bias[n] : 0.0f;
#pragma unroll
        for (int v = 0; v < 8; ++v) {
            const int r = row0 + v + 8 * khalf;
            C[(size_t)r * NDIM + n] = acc[t][v] + bv;
        }
    }
}

// ---------------------------------------------------------------------------
// Per-(node, head) attention scores: e = <hh[n, h, :], a[h, :]>
// ---------------------------------------------------------------------------
__global__ void attn_scores_kernel(const float* __restrict__ hh,
                                   const float* __restrict__ a_src,
                                   const float* __restrict__ a_dst,
                                   float* __restrict__ es,
                                   float* __restrict__ ed) {
    int idx = blockIdx.x * blockDim.x + threadIdx.x;
    if (idx >= N_NODES * HEADS) return;
    const int node = idx >> 2;
    const int h    = idx & 3;
    const float* row = hh + (size_t)node * NDIM + h * HEAD_DIM;
    const float* as  = a_src + h * HEAD_DIM;
    const float* ad  = a_dst + h * HEAD_DIM;
    float s = 0.0f, d = 0.0f;
#pragma unroll
    for (int i = 0; i < HEAD_DIM; ++i) {
        const float v = row[i];
        s += v * as[i];
        d += v * ad[i];
    }
    es[idx] = s;
    ed[idx] = d;
}

__global__ void fill_kernel(float* __restrict__ p, float v, int n) {
    int i = blockIdx.x * blockDim.x + threadIdx.x;
    if (i < n) p[i] = v;
}

// ---------------------------------------------------------------------------
// Edge pass 1: segment max of leaky_relu(e_src[src] + e_dst[dst]) over dst
// ---------------------------------------------------------------------------
__global__ void edge_max_kernel(const int* __restrict__ src,
                                const int* __restrict__ dst,
                                const float* __restrict__ es,
                                const float* __restrict__ ed,
                                float* __restrict__ mx) {
    int idx = blockIdx.x * blockDim.x + threadIdx.x;
    if (idx >= N_EDGES * HEADS) return;
    const int e = idx >> 2, h = idx & 3;
    const int s = src[e], d = dst[e];
    float v = es[s * HEADS + h] + ed[d * HEADS + h];
    v = v > 0.0f ? v : 0.2f * v;
    atomicMaxFloat(&mx[d * HEADS + h], v);
}

// ---------------------------------------------------------------------------
// Edge pass 2: ex = exp(e - m[dst]); denom[dst] += ex
// ---------------------------------------------------------------------------
__global__ void edge_expsum_kernel(const int* __restrict__ src,
                                   const int* __restrict__ dst,
                                   const float* __restrict__ es,
                                   const float* __restrict__ ed,
                                   const float* __restrict__ mx,
                                   float* __restrict__ ex,
                                   float* __restrict__ denom) {
    int idx = blockIdx.x * blockDim.x + threadIdx.x;
    if (idx >= N_EDGES * HEADS) return;
    const int e = idx >> 2, h = idx & 3;
    const int s = src[e], d = dst[e];
    float v = es[s * HEADS + h] + ed[d * HEADS + h];
    v = v > 0.0f ? v : 0.2f * v;
    const float e2 = expf(v - mx[d * HEADS + h]);
    ex[idx] = e2;
    atomicAdd(&denom[d * HEADS + h], e2);
}

// ---------------------------------------------------------------------------
// Edge pass 3: agg[dst, f] += (ex/denom[dst]) * hh[src, f]
// One thread per (edge, feature) -> coalesced reads of hh rows.
// ---------------------------------------------------------------------------
__global__ void edge_aggregate_kernel(const int* __restrict__ src,
                                      const int* __restrict__ dst,
                                      const float* __restrict__ ex,
                                      const float* __restrict__ denom,
                                      const float* __restrict__ hh,
                                      float* __restrict__ agg) {
    int idx = blockIdx.x * blockDim.x + threadIdx.x;
    if (idx >= N_EDGES * NDIM) return;          // 102.4M < INT_MAX
    const int e = idx >> 7;
    const int f = idx & 127;
    const int h = f >> 5;
    const int d = dst[e];
    const float alpha = ex[e * HEADS + h] / denom[d * HEADS + h];
    atomicAdd(&agg[(size_t)d * NDIM + f],
              alpha * hh[(size_t)src[e] * NDIM + f]);
}

// ---------------------------------------------------------------------------
// Global mean pool (sum + counts via atomics)
// ---------------------------------------------------------------------------
__global__ void pool_kernel(const float* __restrict__ agg,
                            const int* __restrict__ batch,
                            float* __restrict__ pooled,
                            float* __restrict__ counts) {
    int idx = blockIdx.x * blockDim.x + threadIdx.x;
    if (idx >= N_NODES * NDIM) return;
    const int node = idx >> 7;
    const int f    = idx & 127;
    const int g    = batch[node];
    atomicAdd(&pooled[g * NDIM + f], agg[idx]);
    if (f == 0) atomicAdd(&counts[g], 1.0f);
}

// ---------------------------------------------------------------------------
// MLP classifier (tiny: 64x128 @ 128x64 relu, then @ 64x2)
// ---------------------------------------------------------------------------
__global__ void mlp1_kernel(const float* __restrict__ pooled,
                            const float* __restrict__ counts,
                            const float* __restrict__ Wc1,
                            const float* __restrict__ bc1,
                            float* __restrict__ hcls) {
    int idx = blockIdx.x * blockDim.x + threadIdx.x;
    if (idx >= N_GRAPHS * 64) return;
    const int g = idx >> 6, j = idx & 63;
    const float inv = 1.0f / fmaxf(counts[g], 1.0f);
    float s = bc1[j];
    for (int k = 0; k < NDIM; ++k)
        s += (pooled[g * NDIM + k] * inv) * Wc1[k * 64 + j];
    hcls[idx] = fmaxf(s, 0.0f);
}

__global__ void mlp2_kernel(const float* __restrict__ hcls,
                            const float* __restrict__ Wc2,
                            const float* __restrict__ bc2,
                            float* __restrict__ out) {
    int idx = blockIdx.x * blockDim.x + threadIdx.x;
    if (idx >= N_GRAPHS * 2) return;
    const int g = idx >> 1, c = idx & 1;
    float s = bc2[c];
    for (int j = 0; j < 64; ++j)
        s += hcls[g * 64 + j] * Wc2[j * 2 + c];
    out[idx] = s;
}

// ---------------------------------------------------------------------------
extern "C" void kernel_launch(void* const* d_in, const int* in_sizes, int n_in,
                              void* d_out, int out_size, void* d_ws, size_t ws_size,
                              hipStream_t stream) {
    const float* x      = (const float*)d_in[0];
    const int*   eidx   = (const int*)d_in[1];
    const int*   batch  = (const int*)d_in[2];
    const float* W_proj = (const float*)d_in[3];
    const float* b_proj = (const float*)d_in[4];
    const float* W_gat  = (const float*)d_in[5];
    const float* a_src  = (const float*)d_in[6];
    const float* a_dst  = (const float*)d_in[7];
    const float* Wc1    = (const float*)d_in[8];
    const float* bc1    = (const float*)d_in[9];
    const float* Wc2    = (const float*)d_in[10];
    const float* bc2    = (const float*)d_in[11];
    float* out = (float*)d_out;

    const int* src = eidx;              // edge_index[0]
    const int* dst = eidx + N_EDGES;    // edge_index[1]

    // workspace layout (fp32)
    float* ws     = (float*)d_ws;
    float* h      = ws;                                   // N*128 (reused as agg)
    float* hh     = h      + (size_t)N_NODES * NDIM;      // N*128
    float* es     = hh     + (size_t)N_NODES * NDIM;      // N*4
    float* ed     = es     + N_NODES * HEADS;             // N*4
    float* mx     = ed     + N_NODES * HEADS;             // N*4
    float* denom  = mx     + N_NODES * HEADS;             // N*4
    float* ex     = denom  + N_NODES * HEADS;             // E*4
    float* pooled = ex     + (size_t)N_EDGES * HEADS;     // 64*128
    float* counts = pooled + N_GRAPHS * NDIM;             // 64
    float* hcls   = counts + N_GRAPHS;                    // 64*64

    const int TILES = N_NODES / 16;                 // 3125 (exact)
    const dim3 blk(256);
    const int gemm_blocks = (TILES + 7) / 8;        // 8 waves per block

    // 1) h = x @ W_proj + b_proj
    hipLaunchKernelGGL(wmma_gemm128, dim3(gemm_blocks), blk, GEMM_LDS_BYTES, stream,
                       x, W_proj, b_proj, h, TILES);
    // 2) hh = h @ W_gat
    hipLaunchKernelGGL(wmma_gemm128, dim3(gemm_blocks), blk, GEMM_LDS_BYTES, stream,
                       h, W_gat, (const float*)nullptr, hh, TILES);
    // 3) per-node attention scores
    hipLaunchKernelGGL(attn_scores_kernel,
                       dim3((N_NODES * HEADS + 255) / 256), blk, 0, stream,
                       hh, a_src, a_dst, es, ed);
    // 4) init buffers (h is dead now -> reuse as aggregation output)
    hipLaunchKernelGGL(fill_kernel, dim3((N_NODES * HEADS + 255) / 256), blk, 0, stream,
                       mx, -INFINITY, N_NODES * HEADS);
    hipLaunchKernelGGL(fill_kernel, dim3((N_NODES * HEADS + 255) / 256), blk, 0, stream,
                       denom, 0.0f, N_NODES * HEADS);
    hipLaunchKernelGGL(fill_kernel, dim3((N_NODES * NDIM + 255) / 256), blk, 0, stream,
                       h, 0.0f, N_NODES * NDIM);
    hipLaunchKernelGGL(fill_kernel, dim3((N_GRAPHS * NDIM + 255) / 256), blk, 0, stream,
                       pooled, 0.0f, N_GRAPHS * NDIM);
    hipLaunchKernelGGL(fill_kernel, dim3(1), blk, 0, stream,
                       counts, 0.0f, N_GRAPHS);
    // 5) segment max over incoming edges
    hipLaunchKernelGGL(edge_max_kernel, dim3((N_EDGES * HEADS + 255) / 256), blk, 0, stream,
                       src, dst, es, ed, mx);
    // 6) exp + denominator
    hipLaunchKernelGGL(edge_expsum_kernel, dim3((N_EDGES * HEADS + 255) / 256), blk, 0, stream,
                       src, dst, es, ed, mx, ex, denom);
    // 7) gather-attend-scatter aggregation (into h, now zeroed)
    hipLaunchKernelGGL(edge_aggregate_kernel, dim3((N_EDGES * NDIM + 255) / 256), blk, 0, stream,
                       src, dst, ex, denom, hh, h);
    // 8) global mean pool
    hipLaunchKernelGGL(pool_kernel, dim3((N_NODES * NDIM + 255) / 256), blk, 0, stream,
                       h, batch, pooled, counts);
    // 9) MLP classifier
    hipLaunchKernelGGL(mlp1_kernel, dim3((N_GRAPHS * 64 + 255) / 256), blk, 0, stream,
                       pooled, counts, Wc1, bc1, hcls);
    hipLaunchKernelGGL(mlp2_kernel, dim3(1), blk, 0, stream,
                       hcls, Wc2, bc2, out);
}